// HungarianMatcher_63754494542406
// MI455X (gfx1250) — compile-verified
//
#include <hip/hip_runtime.h>
#include <hip/hip_bf16.h>

// Problem constants (match reference)
static constexpr int Bn = 8;
static constexpr int Nn = 3000;
static constexpr int Cc = 91;   // classes (also packed LDS row stride; odd -> conflict-free)
static constexpr int Kp = 92;   // padded K for WMMA (multiple of 4)
static constexpr int Tn = 100;  // targets
static constexpr int Tp = 112;  // padded targets (7 x 16 WMMA col tiles)

typedef __attribute__((ext_vector_type(2))) float    v2f;
typedef __attribute__((ext_vector_type(8))) float    v8f;
typedef __attribute__((ext_vector_type(4))) unsigned u32x4;
typedef __attribute__((ext_vector_type(8))) unsigned u32x8;

// ---------------------------------------------------------------------------
// Kernel 1: fused cost matrix.
//   logits tile staged via TDM (tensor_load_to_lds, TENSORcnt-tracked DMA)
//   cost_class via f32 WMMA:  D = P(softmax) x OneHot(labels)
//   cost_bbox (L1) + cost_giou fused in the WMMA epilogue.
// Block: 256 threads (8 wave32s), tile of 128 queries x 112 (padded) targets.
// ---------------------------------------------------------------------------
__global__ __launch_bounds__(256)
void hm_cost_kernel(const float* __restrict__ logits,   // [B,N,C]
                    const float* __restrict__ boxes,    // [B,N,4] cxcywh norm
                    const int*   __restrict__ labels,   // [B,T]
                    const float* __restrict__ tboxes,   // [B,T,4] xyxy px
                    const float* __restrict__ sizes,    // [B,2] (h,w)
                    float* __restrict__ cost,           // [B,N,T]
                    float* __restrict__ costT)          // [B,T,N] or null
{
    constexpr int SA = Cc;                // packed stride 91 (TDM writes rows packed)
    __shared__ float probA[128 * SA + 2]; // softmax probs (+2 slack for K-pad reads)
    __shared__ float qb[128 * 4];         // query boxes cxcywh
    __shared__ float tdat[Tp * 8];        // per target: cx,cy,w,h,x0,y0,x1,y1
    __shared__ int   lab[Tp];             // labels, -1 for pad cols

    const int b    = blockIdx.y;
    const int q0   = blockIdx.x * 128;
    const int tid  = threadIdx.x;
    const int lane = tid & 31;
    const int wave = tid >> 5;

    // ---- TDM: DMA the 91x128 (dword) logits tile into LDS (wave 0 issues) ----
    if (wave == 0) {
        const unsigned long long gaddr =
            (unsigned long long)(const void*)(logits + ((long)b * Nn + q0) * Cc);
        const unsigned ldsA  = (unsigned)(unsigned long long)(const void*)&probA[0];
        const unsigned rows  = (unsigned)(Nn - q0 < 128 ? Nn - q0 : 128); // OOB rows -> 0
        u32x4 g0;
        g0[0] = 1u;                                         // count=1, user D#
        g0[1] = ldsA;                                       // lds_addr
        g0[2] = (unsigned)(gaddr & 0xffffffffull);          // global_addr[31:0]
        g0[3] = (unsigned)((gaddr >> 32) & 0x01ffffffull)   // global_addr[56:32]
                | (2u << 30);                               // type = 2 (image)
        u32x8 g1;
        g1[0] = (2u << 16);                 // workgroup_mask=0, data_size=4B, no flags
        g1[1] = ((unsigned)Cc & 0xffffu) << 16;             // tensor_dim0[15:0]
        g1[2] = (((unsigned)Cc >> 16) & 0xffffu)            // tensor_dim0[31:16]
                | ((rows & 0xffffu) << 16);                 // tensor_dim1[15:0]
        g1[3] = ((rows >> 16) & 0xffffu)                    // tensor_dim1[31:16]
                | ((unsigned)Cc << 16);                     // tile_dim0 = 91
        g1[4] = 128u;                                       // tile_dim1=128, tile_dim2=0
        g1[5] = (unsigned)Cc;                               // tensor_dim0_stride = 91
        g1[6] = 0u;                                         // stride hi / dim1_stride lo
        g1[7] = 0u;
        asm volatile("tensor_load_to_lds %0, %1" :: "s"(g0), "s"(g1) : "memory");
    }

    // ---- stage targets (overlaps with TDM) ----
    const float hs = sizes[b * 2 + 0];
    const float ws = sizes[b * 2 + 1];
    for (int t = tid; t < Tp; t += 256) {
        int L = (t < Tn) ? labels[b * Tn + t] : -1;
        lab[t] = L;
        float x0 = 0.f, y0 = 0.f, x1 = 0.f, y1 = 0.f;
        if (t < Tn) {
            const float* tb = tboxes + ((long)b * Tn + t) * 4;
            x0 = tb[0] / ws; y0 = tb[1] / hs; x1 = tb[2] / ws; y1 = tb[3] / hs;
        }
        tdat[t * 8 + 0] = 0.5f * (x0 + x1);
        tdat[t * 8 + 1] = 0.5f * (y0 + y1);
        tdat[t * 8 + 2] = x1 - x0;
        tdat[t * 8 + 3] = y1 - y0;
        tdat[t * 8 + 4] = x0; tdat[t * 8 + 5] = y0;
        tdat[t * 8 + 6] = x1; tdat[t * 8 + 7] = y1;
    }
    // ---- stage query boxes ----
    for (int idx = tid; idx < 128 * 4; idx += 256) {
        int r = idx >> 2, q = q0 + r;
        qb[idx] = (q < Nn) ? boxes[((long)b * Nn + q) * 4 + (idx & 3)] : 0.0f;
    }
    if (wave == 0) __builtin_amdgcn_s_wait_tensorcnt(0);    // TDM done -> LDS valid
    __syncthreads();

    // ---- softmax per query row (one thread per row; odd LDS stride) ----
    if (tid < 128) {
        float m = -__builtin_inff();
        for (int c = 0; c < Cc; c++) m = fmaxf(m, probA[tid * SA + c]);
        float s = 0.f;
        for (int c = 0; c < Cc; c++) {
            float e = __expf(probA[tid * SA + c] - m);
            probA[tid * SA + c] = e;
            s += e;
        }
        float inv = 1.0f / s;
        for (int c = 0; c < Cc; c++) probA[tid * SA + c] *= inv;
        if (tid == 0) { probA[128 * SA] = 0.f; probA[128 * SA + 1] = 0.f; } // K-pad slack
    }
    __syncthreads();

    // ---- WMMA: 8 row-tiles x 7 col-tiles = 56 output tiles, 7 per wave ----
    // K runs to 92; column 91 reads the neighbor row's finite prob (or zeroed
    // slack for row 127) and is multiplied by a guaranteed-zero one-hot entry.
    const int half = lane >> 4;     // 0: K pair {0,1}, 1: K pair {2,3}
    const int lm   = lane & 15;
    for (int tile = wave; tile < 56; tile += 8) {
        const int tr = tile & 7;          // row tile
        const int tc = tile >> 3;         // col tile
        const int rbase = tr * 16;
        const int cbase = tc * 16;
        const int col   = cbase + lm;     // this lane's N column (B/D layout)
        const int labn  = lab[col];       // -1 on pad -> zero one-hot

        v8f acc = {0.f, 0.f, 0.f, 0.f, 0.f, 0.f, 0.f, 0.f};
        #pragma unroll
        for (int k0 = 0; k0 < Kp; k0 += 4) {
            const int ks = k0 + half * 2;
            v2f a, bb;
            a.x = probA[(rbase + lm) * SA + ks];
            a.y = probA[(rbase + lm) * SA + ks + 1];
            bb.x = (labn == ks)     ? 1.0f : 0.0f;   // one-hot column, built in-regs
            bb.y = (labn == ks + 1) ? 1.0f : 0.0f;
            acc = __builtin_amdgcn_wmma_f32_16x16x4_f32(
                false, a, false, bb, (short)0, acc, false, false);
        }

        // ---- epilogue: fuse L1 + GIoU, write cost (and transposed copy) ----
        const float tcx = tdat[col * 8 + 0], tcy = tdat[col * 8 + 1];
        const float tw  = tdat[col * 8 + 2], th  = tdat[col * 8 + 3];
        const float tx0 = tdat[col * 8 + 4], ty0 = tdat[col * 8 + 5];
        const float tx1 = tdat[col * 8 + 6], ty1 = tdat[col * 8 + 7];
        const float areaT = (tx1 - tx0) * (ty1 - ty0);
        #pragma unroll
        for (int r = 0; r < 8; r++) {
            const int row = rbase + r + 8 * half;   // C/D VGPR layout
            const int q = q0 + row;
            if (q < Nn && col < Tn) {
                const float qcx = qb[row * 4 + 0], qcy = qb[row * 4 + 1];
                const float qw  = qb[row * 4 + 2], qh  = qb[row * 4 + 3];
                const float l1 = fabsf(qcx - tcx) + fabsf(qcy - tcy) +
                                 fabsf(qw - tw)   + fabsf(qh - th);
                const float qx0 = qcx - 0.5f * qw, qy0 = qcy - 0.5f * qh;
                const float qx1 = qcx + 0.5f * qw, qy1 = qcy + 0.5f * qh;
                const float areaQ = (qx1 - qx0) * (qy1 - qy0);
                const float ix0 = fmaxf(qx0, tx0), iy0 = fmaxf(qy0, ty0);
                const float ix1 = fminf(qx1, tx1), iy1 = fminf(qy1, ty1);
                const float iw = fmaxf(ix1 - ix0, 0.f), ih = fmaxf(iy1 - iy0, 0.f);
                const float inter = iw * ih;
                const float uni = areaQ + areaT - inter;
                const float iou = inter / uni;
                const float ex0 = fminf(qx0, tx0), ey0 = fminf(qy0, ty0);
                const float ex1 = fmaxf(qx1, tx1), ey1 = fmaxf(qy1, ty1);
                const float areaE = fmaxf(ex1 - ex0, 0.f) * fmaxf(ey1 - ey0, 0.f);
                const float giou = iou - (areaE - uni) / areaE;
                const float cclass = -acc[r];
                const float cval = 2.0f * cclass + 5.0f * l1 + 2.0f * (-giou);
                cost[((long)b * Nn + q) * Tn + col] = cval;
                if (costT) costT[((long)b * Tn + col) * Nn + q] = cval;
            }
        }
    }
}

// ---------------------------------------------------------------------------
// Kernel 2: exact rectangular LSA (shortest augmenting path with duals),
// one workgroup (1024 threads = 32 wave32s) per batch. Solver state in LDS.
// Argmin: strided scan -> wave32 shuffle min -> 32-entry LDS fold (2 barriers).
// ---------------------------------------------------------------------------
__global__ __launch_bounds__(1024)
void hm_lsa_kernel(const float* __restrict__ cost,    // [B,N,T]
                   const float* __restrict__ costT,   // [B,T,N] or null
                   float* __restrict__ out)           // full output buffer
{
    constexpr int NP = 3008;                          // padded columns
    __shared__ float shortest[NP];
    __shared__ float vdual[NP];
    __shared__ int   pathj[NP];
    __shared__ int   row4col[NP];
    __shared__ unsigned char SC[NP];
    __shared__ float u[128];
    __shared__ int   col4row[128];
    __shared__ unsigned char SR[128];
    __shared__ float redV[32];
    __shared__ int   redI[32];
    __shared__ int   ctrl[2];        // [0] = current row i, [1] = sink col
    __shared__ float minv[1];

    const int b    = blockIdx.x;
    const int tid  = threadIdx.x;
    const int lane = tid & 31;
    const int wave = tid >> 5;
    const int NTH  = blockDim.x;     // 1024 = 32 wave32s
    const float INF = __builtin_inff();

    for (int j = tid; j < Nn; j += NTH) { vdual[j] = 0.f; row4col[j] = -1; }
    for (int i = tid; i < Tn; i += NTH) { u[i] = 0.f; col4row[i] = -1; }
    __syncthreads();

    for (int cur = 0; cur < Tn; cur++) {
        for (int j = tid; j < Nn; j += NTH) { shortest[j] = INF; SC[j] = 0; pathj[j] = -1; }
        for (int i = tid; i < Tn; i += NTH) SR[i] = 0;
        if (tid == 0) { ctrl[0] = cur; ctrl[1] = -1; minv[0] = 0.f; }
        __syncthreads();

        while (true) {
            const int i = ctrl[0];
            const float minVal = minv[0];
            if (tid == 0) SR[i] = 1;
            const float ui = u[i];
            const float* rowp = costT ? (costT + ((long)b * Tn + i) * Nn) : nullptr;

            // relax: shortest path distances over unvisited columns
            for (int j = tid; j < Nn; j += NTH) {
                if (rowp && j + NTH < Nn)
                    __builtin_prefetch(rowp + j + NTH, 0, 1);   // global_prefetch_b8
                if (!SC[j]) {
                    const float cij = rowp ? rowp[j]
                                           : cost[((long)b * Nn + j) * Tn + i];
                    const float d = minVal + cij - ui - vdual[j];
                    if (d < shortest[j]) { shortest[j] = d; pathj[j] = i; }
                }
            }
            __syncthreads();

            // argmin over unvisited columns, first-occurrence tie-break.
            float lv = INF; int li = -1;
            for (int j = tid; j < Nn; j += NTH) {
                const float sv = SC[j] ? INF : shortest[j];
                if (sv < lv || (sv == lv && (unsigned)j < (unsigned)li)) { lv = sv; li = j; }
            }
            #pragma unroll
            for (int off = 16; off > 0; off >>= 1) {           // wave32 shuffle min
                const float ov = __shfl_xor(lv, off, 32);
                const int   oi = __shfl_xor(li, off, 32);
                if (ov < lv || (ov == lv && (unsigned)oi < (unsigned)li)) { lv = ov; li = oi; }
            }
            if (lane == 0) { redV[wave] = lv; redI[wave] = li; }
            __syncthreads();
            if (tid < 32) {                                    // fold 32 wave results
                lv = redV[tid]; li = redI[tid];
                #pragma unroll
                for (int off = 16; off > 0; off >>= 1) {
                    const float ov = __shfl_xor(lv, off, 32);
                    const int   oi = __shfl_xor(li, off, 32);
                    if (ov < lv || (ov == lv && (unsigned)oi < (unsigned)li)) { lv = ov; li = oi; }
                }
                if (tid == 0) {
                    const int j = li;
                    minv[0] = lv;
                    SC[j] = 1;
                    if (row4col[j] == -1) ctrl[1] = j;
                    else                  ctrl[0] = row4col[j];
                }
            }
            __syncthreads();
            if (ctrl[1] >= 0) break;
        }

        // dual updates (reference order: before augmentation)
        const float minVal = minv[0];
        if (tid == 0) u[cur] += minVal;
        for (int i = tid; i < Tn; i += NTH)
            if (SR[i] && i != cur) u[i] += minVal - shortest[col4row[i]];
        for (int j = tid; j < Nn; j += NTH)
            if (SC[j]) vdual[j] -= minVal - shortest[j];
        __syncthreads();

        // augment along the alternating path (serial, thread 0)
        if (tid == 0) {
            int j = ctrl[1];
            while (true) {
                const int i = pathj[j];
                row4col[j] = i;
                const int tmp = col4row[i];
                col4row[i] = j;
                j = tmp;
                if (i == cur) break;
            }
        }
        __syncthreads();
    }

    // Emit src/tgt in scipy order: order = argsort(col4row) (values distinct).
    if (tid == 0) {
        float* srcOut = out + (long)Bn * Nn * Tn + (long)b * Tn;
        float* tgtOut = out + (long)Bn * Nn * Tn + (long)Bn * Tn + (long)b * Tn;
        for (int i = 0; i < Tn; i++) {
            const int ci = col4row[i];
            int rank = 0;
            for (int k = 0; k < Tn; k++) rank += (col4row[k] < ci);
            srcOut[rank] = (float)ci;
            tgtOut[rank] = (float)i;
        }
    }
}

// ---------------------------------------------------------------------------
extern "C" void kernel_launch(void* const* d_in, const int* in_sizes, int n_in,
                              void* d_out, int out_size, void* d_ws, size_t ws_size,
                              hipStream_t stream) {
    const float* logits = (const float*)d_in[0];   // [B,N,C]
    const float* boxes  = (const float*)d_in[1];   // [B,N,4]
    const int*   labels = (const int*)d_in[2];     // [B,T]
    const float* tboxes = (const float*)d_in[3];   // [B,T,4]
    const float* sizes  = (const float*)d_in[4];   // [B,2]
    float* out = (float*)d_out;

    const size_t needT = (size_t)Bn * Tn * Nn * sizeof(float);
    float* costT = (ws_size >= needT) ? (float*)d_ws : nullptr;

    dim3 g1((Nn + 127) / 128, Bn);
    hm_cost_kernel<<<g1, 256, 0, stream>>>(logits, boxes, labels, tboxes, sizes,
                                           out, costT);
    hm_lsa_kernel<<<Bn, 1024, 0, stream>>>(out, costT, out);
}